// GCNLayer_89996744720583
// MI455X (gfx1250) — compile-verified
//
#include <hip/hip_runtime.h>
#include <hip/hip_bf16.h>

typedef float v2f __attribute__((ext_vector_type(2)));
typedef float v8f __attribute__((ext_vector_type(8)));

#define GCN_D 128

// ---------------------------------------------------------------------------
// GEMM: HW[n x 128] = H[n x 128] @ W[128 x 128], fp32 WMMA 16x16x4.
// Block = 256 threads = 8 waves; block covers 128 rows; wave covers 16 rows
// x all 128 cols (8 accumulator tiles). W staged in LDS (64 KB) via the
// CDNA5 async-to-LDS path when available.
// A layout (16x4 fp32): lanes 0-15 hold M=0..15; VGPR0 = K0|K2, VGPR1 = K1|K3.
// B layout (4x16 fp32): VGPR0 = row K0 (lanes0-15) | K2 (lanes16-31); VGPR1 = K1|K3.
// C/D layout (16x16 f32): VGPR r -> M=r (lanes0-15), M=r+8 (lanes16-31), N=lane%16.
// OOB rows: A row index is clamped (no branch); a garbage A row only affects
// the same output row M, which is never stored.
// ---------------------------------------------------------------------------
__global__ __launch_bounds__(256) void k_gemm_wmma(
    const float* __restrict__ H, const float* __restrict__ W,
    float* __restrict__ HW, int nrows) {
  __shared__ float Wl[GCN_D * GCN_D];
  const int tid = threadIdx.x;

#if defined(__gfx1250__) && __has_builtin(__builtin_amdgcn_global_load_async_to_lds_b128)
  {
    typedef int v4i32 __attribute__((vector_size(16)));
    typedef __attribute__((address_space(1))) v4i32* gv4p;   // AS1 ("__device__")
    typedef __attribute__((address_space(3))) v4i32* lv4p;   // AS3 ("__shared__")
    for (int i = tid; i < (GCN_D * GCN_D) / 4; i += 256) {
      gv4p g = (gv4p)(uintptr_t)(W + 4 * i);
      lv4p l = (lv4p)(unsigned)(uintptr_t)(&Wl[4 * i]);
      __builtin_amdgcn_global_load_async_to_lds_b128(g, l, 0, 0);
    }
#if __has_builtin(__builtin_amdgcn_s_wait_asynccnt)
    __builtin_amdgcn_s_wait_asynccnt(0);
#else
    asm volatile("s_wait_asynccnt 0x0" ::: "memory");
#endif
  }
#else
  for (int i = tid; i < (GCN_D * GCN_D) / 4; i += 256)
    reinterpret_cast<float4*>(Wl)[i] = reinterpret_cast<const float4*>(W)[i];
#endif
  __syncthreads();

  const int wave  = tid >> 5;
  const int lane  = tid & 31;
  const int lhalf = lane >> 4;   // 0: K even pair, 1: K odd pair
  const int l16   = lane & 15;
  const int rowBase = blockIdx.x * 128 + wave * 16;
  const int arow    = rowBase + l16;
  const int arowC   = (arow < nrows) ? arow : (nrows - 1);   // clamp, branch-free
  const float2* Arow = reinterpret_cast<const float2*>(H + (size_t)arowC * GCN_D);

  v8f acc[8];
#pragma unroll
  for (int n = 0; n < 8; ++n) { v8f z = {}; acc[n] = z; }

#pragma unroll 4
  for (int k = 0; k < GCN_D; k += 4) {
    const float2 t = Arow[(k >> 1) + lhalf];   // elems (k+2*lhalf, k+2*lhalf+1)
    v2f a;
    a.x = t.x; a.y = t.y;
    const float* bp = &Wl[(k + 2 * lhalf) * GCN_D + l16];
#pragma unroll
    for (int n = 0; n < 8; ++n) {
      v2f b;
      b.x = bp[n * 16];
      b.y = bp[GCN_D + n * 16];
      acc[n] = __builtin_amdgcn_wmma_f32_16x16x4_f32(
          false, a, false, b, (short)0, acc[n], false, false);
    }
  }

#pragma unroll
  for (int n = 0; n < 8; ++n) {
#pragma unroll
    for (int r = 0; r < 8; ++r) {
      const int row = rowBase + r + 8 * lhalf;
      if (row < nrows)
        HW[(size_t)row * GCN_D + n * 16 + l16] = acc[n][r];
    }
  }
}

// ---------------------------------------------------------------------------
// Degree kernels
// ---------------------------------------------------------------------------
__global__ void k_zero_u32(unsigned* __restrict__ p, int n) {
  const int i = blockIdx.x * blockDim.x + threadIdx.x;
  if (i < n) p[i] = 0u;
}

__global__ void k_count_deg(const int* __restrict__ dst,
                            unsigned* __restrict__ degi, int E) {
  const int e = blockIdx.x * blockDim.x + threadIdx.x;
  if (e < E) atomicAdd(&degi[dst[e]], 1u);
}

__global__ void k_fin_deg(const unsigned* __restrict__ degi,
                          float* __restrict__ dinv,
                          float* __restrict__ winv, int n) {
  const int i = blockIdx.x * blockDim.x + threadIdx.x;
  if (i < n) {
    const float d = (float)degi[i] + 2.0f;   // improved=True self-loop weight 2
    dinv[i] = rsqrtf(d);
    winv[i] = 2.0f / d;
  }
}

// ---------------------------------------------------------------------------
// agg[i][c] = winv[i] * hw[i][c] + bias[c]   (self-loop term + bias init)
// ---------------------------------------------------------------------------
__global__ __launch_bounds__(256) void k_selfloop(
    const float* __restrict__ hw, const float* __restrict__ winv,
    const float* __restrict__ bias, float* __restrict__ agg, int n) {
  const int idx = blockIdx.x * blockDim.x + threadIdx.x;   // one float4 per thread
  if (idx >= n * 32) return;
  const int row = idx >> 5;
  const int c4  = (idx & 31) << 2;
  const float  w = winv[row];
  const float4 v = *reinterpret_cast<const float4*>(hw + (size_t)row * GCN_D + c4);
  const float4 b = *reinterpret_cast<const float4*>(bias + c4);
  float4 o;
  o.x = v.x * w + b.x; o.y = v.y * w + b.y;
  o.z = v.z * w + b.z; o.w = v.w * w + b.w;
  *reinterpret_cast<float4*>(agg + (size_t)row * GCN_D + c4) = o;
}

// ---------------------------------------------------------------------------
// One wave per edge: agg[dst] += hw[src] * (dinv[src]*dinv[dst])
// Edge index is forced into an SGPR (readfirstlane) so src/dst/dinv reads
// become scalar loads; the 512B row gather is fully coalesced; fp32 atomic
// scatter stays resident in L2 (hw/agg are 25.6MB << 192MB).
// ---------------------------------------------------------------------------
__global__ __launch_bounds__(256) void k_edge_scatter(
    const int* __restrict__ src, const int* __restrict__ dst,
    const float* __restrict__ dinv, const float* __restrict__ hw,
    float* __restrict__ agg, int E) {
  const int e = __builtin_amdgcn_readfirstlane(blockIdx.x * 8 + (threadIdx.x >> 5));
  if (e >= E) return;                      // wave-uniform branch
  const int lane = threadIdx.x & 31;
  const int s = src[e];
  const int d = dst[e];
  const float norm = dinv[s] * dinv[d];
  const float4 v = *reinterpret_cast<const float4*>(hw + (size_t)s * GCN_D + lane * 4);
  float* ap = agg + (size_t)d * GCN_D + lane * 4;
  atomicAdd(ap + 0, v.x * norm);
  atomicAdd(ap + 1, v.y * norm);
  atomicAdd(ap + 2, v.z * norm);
  atomicAdd(ap + 3, v.w * norm);
}

// ---------------------------------------------------------------------------
// BatchNorm statistics + apply
// ---------------------------------------------------------------------------
__global__ void k_zero_f32(float* __restrict__ p, int n) {
  const int i = blockIdx.x * blockDim.x + threadIdx.x;
  if (i < n) p[i] = 0.0f;
}

__global__ __launch_bounds__(256) void k_bn_stats(
    const float* __restrict__ agg, float* __restrict__ sum,
    float* __restrict__ sumsq, int n) {
  __shared__ float ssum[256];
  __shared__ float ssq[256];
  const int t = threadIdx.x;
  const int c = t & 127;
  const int half = t >> 7;
  float s = 0.0f, q = 0.0f;
  for (int r = blockIdx.x * 2 + half; r < n; r += gridDim.x * 2) {
    const float v = agg[(size_t)r * GCN_D + c];
    s += v;
    q += v * v;
  }
  ssum[t] = s; ssq[t] = q;
  __syncthreads();
  if (t < 128) {
    atomicAdd(&sum[c],   ssum[t] + ssum[t + 128]);
    atomicAdd(&sumsq[c], ssq[t]  + ssq[t + 128]);
  }
}

__global__ void k_bn_final(const float* __restrict__ sum,
                           const float* __restrict__ sumsq,
                           float* __restrict__ mean,
                           float* __restrict__ istd, float invn) {
  const int c = threadIdx.x;
  const float m = sum[c] * invn;
  const float v = sumsq[c] * invn - m * m;   // biased variance (training mode)
  mean[c] = m;
  istd[c] = rsqrtf(v + 1e-5f);
}

__global__ __launch_bounds__(256) void k_bn_apply(
    const float* __restrict__ agg, const float* __restrict__ mean,
    const float* __restrict__ istd, const float* __restrict__ g,
    const float* __restrict__ be, float* __restrict__ out, int n) {
  const int idx = blockIdx.x * blockDim.x + threadIdx.x;
  if (idx >= n * 32) return;
  const int row = idx >> 5;
  const int c4  = (idx & 31) << 2;
  const float4 v  = *reinterpret_cast<const float4*>(agg  + (size_t)row * GCN_D + c4);
  const float4 m  = *reinterpret_cast<const float4*>(mean + c4);
  const float4 is = *reinterpret_cast<const float4*>(istd + c4);
  const float4 gg = *reinterpret_cast<const float4*>(g    + c4);
  const float4 bb = *reinterpret_cast<const float4*>(be   + c4);
  float4 o;
  o.x = fmaxf((v.x - m.x) * is.x * gg.x + bb.x, 0.0f);
  o.y = fmaxf((v.y - m.y) * is.y * gg.y + bb.y, 0.0f);
  o.z = fmaxf((v.z - m.z) * is.z * gg.z + bb.z, 0.0f);
  o.w = fmaxf((v.w - m.w) * is.w * gg.w + bb.w, 0.0f);
  *reinterpret_cast<float4*>(out + (size_t)row * GCN_D + c4) = o;
}

// ---------------------------------------------------------------------------
// Host-side launcher (graph-capture safe: all launches on `stream`, scratch
// only from d_ws, everything we read is initialized here).
// ---------------------------------------------------------------------------
extern "C" void kernel_launch(void* const* d_in, const int* in_sizes, int n_in,
                              void* d_out, int out_size, void* d_ws, size_t ws_size,
                              hipStream_t stream) {
  const float* x  = (const float*)d_in[0];
  const int*   ei = (const int*)d_in[1];
  const int n = in_sizes[0] / GCN_D;   // 50000
  const int E = in_sizes[1] / 2;       // 1600000
  const int* src = ei;
  const int* dst = ei + E;

  char* ws = (char*)d_ws;
  const size_t matBytes = (size_t)n * GCN_D * sizeof(float);
  float*    hw      = (float*)(ws);
  float*    agg     = (float*)(ws + matBytes);
  float*    hcur    = (float*)(ws + 2 * matBytes);
  float*    dinv    = (float*)(ws + 3 * matBytes);
  float*    winv    = dinv + n;
  unsigned* degi    = (unsigned*)(winv + n);
  float*    bnsum   = (float*)(degi + n);
  float*    bnsumsq = bnsum + GCN_D;
  float*    bnmean  = bnsumsq + GCN_D;
  float*    bnistd  = bnmean + GCN_D;

  const int TB = 256;

  k_zero_u32 <<<(n + TB - 1) / TB, TB, 0, stream>>>(degi, n);
  k_count_deg<<<(E + TB - 1) / TB, TB, 0, stream>>>(dst, degi, E);
  k_fin_deg  <<<(n + TB - 1) / TB, TB, 0, stream>>>(degi, dinv, winv, n);

  const float* hin = x;
  for (int layer = 0; layer < 3; ++layer) {
    const float* Wp  = (const float*)d_in[2 + 4 * layer];
    const float* bp  = (const float*)d_in[3 + 4 * layer];
    const float* gp  = (const float*)d_in[4 + 4 * layer];
    const float* bep = (const float*)d_in[5 + 4 * layer];
    float* hout = (layer == 2) ? (float*)d_out : hcur;

    k_gemm_wmma   <<<(n + 127) / 128,          TB, 0, stream>>>(hin, Wp, hw, n);
    k_selfloop    <<<((n * 32) + TB - 1) / TB, TB, 0, stream>>>(hw, winv, bp, agg, n);
    k_edge_scatter<<<(E + 7) / 8,              TB, 0, stream>>>(src, dst, dinv, hw, agg, E);
    k_zero_f32    <<<1, 256, 0, stream>>>(bnsum, 2 * GCN_D);
    k_bn_stats    <<<1024, TB, 0, stream>>>(agg, bnsum, bnsumsq, n);
    k_bn_final    <<<1, GCN_D, 0, stream>>>(bnsum, bnsumsq, bnmean, bnistd, 1.0f / (float)n);
    k_bn_apply    <<<((n * 32) + TB - 1) / TB, TB, 0, stream>>>(agg, bnmean, bnistd, gp, bep, hout, n);
    hin = hout;
  }
}